// MCUDetectionLoss_24584392802460
// MI455X (gfx1250) — compile-verified
//
#include <hip/hip_runtime.h>
#include <math.h>

// ---------------- problem constants (from reference setup_inputs) -----------
#define NC          24
#define ALPHA_F     0.25f
#define BATCH       32
#define H4C         128
#define W4C         128
#define H5C         64
#define W5C         64
#define T4C         200
#define T5C         100

// target-pass block counts (256 threads each)
#define NB_T4       25      // 32*200/256
#define NB_T5       13      // ceil(32*100/256)
#define SLOT5       NB_T4
#define NPT         (NB_T4 + NB_T5)   // 38
// dense-pass block counts
#define NB_D4       128
#define NB_D5       32

// workspace float-section layout (offsets in floats)
#define P_BBOX      0
#define P_OBJP      64
#define P_CLS       128
#define P_NB4       192
#define P_NC4       320
#define P_NB5       448
#define P_NC5       480
#define WS_FLOATS   512
#define MASK4_BYTES ((size_t)BATCH * H4C * W4C)   // 524288
#define MASK5_BYTES ((size_t)BATCH * H5C * W5C)   // 131072

typedef __attribute__((ext_vector_type(2))) float v2f;
typedef __attribute__((ext_vector_type(8))) float v8f;

// ---------------- wave32 half-sums via V_WMMA_F32_16X16X4_F32 ---------------
// A(16x4): lanes 0-15 hold {K0,K1}, lanes 16-31 hold {K2,K3}. With a={v,0},
// B=ones: D[m][n] = v[m] + v[m+16] (n-independent). D layout: VGPR r,
// lanes 0-15 -> M=r, lanes 16-31 -> M=8+r. Summing c[0..7] in-lane leaves
// sum(v[0..15]+v[16..31] | m<8) replicated in lanes 0-15 and the m>=8 half
// replicated in lanes 16-31: lane 0 and lane 16 carry the two half-sums,
// no cross-lane shuffle needed.
__device__ __forceinline__ float wave_half_sum_wmma(float v) {
    v2f a; a[0] = v;    a[1] = 0.0f;
    v2f b; b[0] = 1.0f; b[1] = 1.0f;
    v8f c = {};
    c = __builtin_amdgcn_wmma_f32_16x16x4_f32(
            /*neg_a=*/false, a, /*neg_b=*/false, b,
            /*c_mod=*/(short)0, c, /*reuse_a=*/false, /*reuse_b=*/false);
    return ((c[0] + c[1]) + (c[2] + c[3])) + ((c[4] + c[5]) + (c[6] + c[7]));
}

// stage one quantity's half-sums into a 16-slot LDS region (256-thread block)
__device__ __forceinline__ void block_stage(float v, float* slots16) {
    const float s = wave_half_sum_wmma(v);
    const int t = threadIdx.x;
    if ((t & 15) == 0) slots16[t >> 4] = s;   // lanes 0 and 16 of each wave
}

__device__ __forceinline__ float fold16(const float* slots16) {
    float r = 0.0f;
    #pragma unroll
    for (int i = 0; i < 16; ++i) r += slots16[i];
    return r;
}

__device__ __forceinline__ float bce_logits(float x, float z) {
    return fmaxf(x, 0.0f) - x * z + log1pf(expf(-fabsf(x)));
}

// ---------------- kernel: zero the mask bytes -------------------------------
__global__ void zero_words(unsigned int* __restrict__ p, unsigned int nwords) {
    const unsigned int stride = gridDim.x * blockDim.x;
    for (unsigned int i = blockIdx.x * blockDim.x + threadIdx.x; i < nwords; i += stride)
        p[i] = 0u;
}

// ---------------- kernel: per-target gather losses + positive mask ----------
template <int T, int H, int W>
__global__ void target_pass(const float* __restrict__ cls_pred,
                            const float* __restrict__ reg_pred,
                            const float* __restrict__ targets,
                            unsigned char* __restrict__ mask,
                            float* __restrict__ p_bbox,
                            float* __restrict__ p_objp,
                            float* __restrict__ p_cls,
                            int slot_off) {
    __shared__ float smem[3 * 16];
    const int i = blockIdx.x * blockDim.x + threadIdx.x;
    constexpr int total = BATCH * T;
    constexpr size_t HW = (size_t)H * W;
    float lb = 0.0f, lo = 0.0f, lc = 0.0f;
    if (i < total) {
        const int b = i / T;
        const float* tg = targets + (size_t)i * 6;
        const int   tcls = (int)tg[0];
        const float tx = tg[1] * (float)W;
        const float ty = tg[2] * (float)H;
        const float tw = tg[3] * (float)W;
        const float th = tg[4] * (float)H;
        const int gx = min(max((int)tx, 0), W - 1);
        const int gy = min(max((int)ty, 0), H - 1);
        const size_t cell = (size_t)gy * W + gx;

        // reg gather: (B,4,H,W)
        const float* rp = reg_pred + (size_t)b * 4 * HW + cell;
        const float r0 = rp[0], r1 = rp[HW], r2 = rp[2 * HW], r3 = rp[3 * HW];
        const float px = fminf(fmaxf((float)gx + r0, 0.0f), (float)W);
        const float py = fminf(fmaxf((float)gy + r1, 0.0f), (float)H);
        const float pw = fminf(fmaxf(expf(r2), 1.0f), (float)W) * 0.1f;
        const float ph = fminf(fmaxf(expf(r3), 1.0f), (float)H) * 0.1f;
        const float pb[4] = {px - pw * 0.5f, py - ph * 0.5f, px + pw * 0.5f, py + ph * 0.5f};
        const float tb[4] = {tx - tw * 0.5f, ty - th * 0.5f, tx + tw * 0.5f, ty + th * 0.5f};
        float s = 0.0f;
        #pragma unroll
        for (int k = 0; k < 4; ++k) {
            const float d = pb[k] - tb[k];
            const float ad = fabsf(d);
            s += (ad < 1.0f) ? (0.5f * d * d) : (ad - 0.5f);
        }
        lb = s * 0.25f;   // mean over 4 coords

        // obj + focal cls gather: (B,1+NC,H,W)
        const float* cp = cls_pred + (size_t)b * (1 + NC) * HW + cell;
        const float ox = cp[0];
        lo = bce_logits(ox, 1.0f);
        float fsum = 0.0f;
        #pragma unroll 4
        for (int c = 0; c < NC; ++c) {
            const float x  = cp[(size_t)(c + 1) * HW];
            const float oh = (tcls == c && tcls < NC) ? 1.0f : 0.0f;
            const float bce = bce_logits(x, oh);
            const float p  = 1.0f / (1.0f + expf(-x));
            const float pt = p * oh + (1.0f - p) * (1.0f - oh);
            const float om = 1.0f - pt;
            fsum += ALPHA_F * om * om * bce;
        }
        lc = fsum * (1.0f / (float)NC);

        mask[(size_t)b * HW + cell] = 1;   // duplicate writes all store 1: benign
    }
    block_stage(lb, smem + 0);
    block_stage(lo, smem + 16);
    block_stage(lc, smem + 32);
    __syncthreads();
    if (threadIdx.x == 0) {
        p_bbox[slot_off + blockIdx.x] = fold16(smem + 0);
        p_objp[slot_off + blockIdx.x] = fold16(smem + 16);
        p_cls [slot_off + blockIdx.x] = fold16(smem + 32);
    }
}

// ---------------- kernel: dense negative-obj BCE over channel 0 -------------
template <int H, int W>
__global__ void dense_pass(const float* __restrict__ cls_pred,
                           const unsigned char* __restrict__ mask,
                           float* __restrict__ p_nb,
                           float* __restrict__ p_nc) {
    __shared__ float smem[2 * 16];
    constexpr size_t HW     = (size_t)H * W;      // power of two
    constexpr size_t HW4    = HW / 4;
    constexpr size_t total4 = (size_t)BATCH * HW4;
    const size_t stride = (size_t)gridDim.x * blockDim.x;
    float nb = 0.0f, nc = 0.0f;
    for (size_t idx = (size_t)blockIdx.x * blockDim.x + threadIdx.x; idx < total4; idx += stride) {
        const size_t b   = idx / HW4;             // shift (compile-time pow2)
        const size_t rem = (idx - b * HW4) * 4;
        const float* xp  = cls_pred + b * (size_t)(1 + NC) * HW + rem;
        __builtin_prefetch(xp + stride * 4, 0, 0);   // global_prefetch_b8
        const float4 x4 = *(const float4*)xp;        // global_load_b128
        const unsigned int m4 = *(const unsigned int*)(mask + idx * 4);
        const float xs[4] = {x4.x, x4.y, x4.z, x4.w};
        #pragma unroll
        for (int k = 0; k < 4; ++k) {
            const float w = ((m4 >> (8 * k)) & 0xffu) ? 0.0f : 1.0f;
            nb += w * (fmaxf(xs[k], 0.0f) + log1pf(expf(-fabsf(xs[k]))));  // bce(x,0)
            nc += w;
        }
    }
    block_stage(nb, smem + 0);
    block_stage(nc, smem + 16);
    __syncthreads();
    if (threadIdx.x == 0) {
        p_nb[blockIdx.x] = fold16(smem + 0);
        p_nc[blockIdx.x] = fold16(smem + 16);
    }
}

// ---------------- kernel: deterministic final combine -----------------------
__global__ void final_pass(const float* __restrict__ ws, float* __restrict__ out) {
    if (threadIdx.x != 0 || blockIdx.x != 0) return;
    float bb = 0.0f, op = 0.0f, cl = 0.0f;
    for (int i = 0; i < NPT; ++i)  bb += ws[P_BBOX + i];
    for (int i = 0; i < NPT; ++i)  op += ws[P_OBJP + i];
    for (int i = 0; i < NPT; ++i)  cl += ws[P_CLS  + i];
    float nb4 = 0.0f, nc4 = 0.0f, nb5 = 0.0f, nc5 = 0.0f;
    for (int i = 0; i < NB_D4; ++i) nb4 += ws[P_NB4 + i];
    for (int i = 0; i < NB_D4; ++i) nc4 += ws[P_NC4 + i];
    for (int i = 0; i < NB_D5; ++i) nb5 += ws[P_NB5 + i];
    for (int i = 0; i < NB_D5; ++i) nc5 += ws[P_NC5 + i];

    const float n = (float)(BATCH * (T4C + T5C));                // 9600
    const float loss_bbox = bb / n;
    const float loss_cls  = cl / n;
    const float lo_sum = op + (nb4 / nc4) * 0.5f + (nb5 / nc5) * 0.5f;
    const float total_grid = (float)((size_t)BATCH * H4C * W4C + (size_t)BATCH * H5C * W5C);
    const float loss_obj = lo_sum / (total_grid + 1e-8f);
    out[0] = loss_bbox + loss_obj + loss_cls;
    out[1] = loss_bbox;
    out[2] = loss_obj;
    out[3] = loss_cls;
}

// ---------------- launcher --------------------------------------------------
extern "C" void kernel_launch(void* const* d_in, const int* in_sizes, int n_in,
                              void* d_out, int out_size, void* d_ws, size_t ws_size,
                              hipStream_t stream) {
    (void)in_sizes; (void)n_in; (void)out_size; (void)ws_size;
    const float* cls4 = (const float*)d_in[0];
    const float* reg4 = (const float*)d_in[1];
    const float* cls5 = (const float*)d_in[2];
    const float* reg5 = (const float*)d_in[3];
    const float* tg4  = (const float*)d_in[4];
    const float* tg5  = (const float*)d_in[5];

    float* ws = (float*)d_ws;
    unsigned char* mask4 = (unsigned char*)d_ws + WS_FLOATS * sizeof(float);
    unsigned char* mask5 = mask4 + MASK4_BYTES;

    // zero both masks every call (harness poisons ws once, never restores)
    const unsigned int nwords = (unsigned int)((MASK4_BYTES + MASK5_BYTES) / 4);
    zero_words<<<160, 256, 0, stream>>>((unsigned int*)mask4, nwords);

    target_pass<T4C, H4C, W4C><<<NB_T4, 256, 0, stream>>>(
        cls4, reg4, tg4, mask4, ws + P_BBOX, ws + P_OBJP, ws + P_CLS, 0);
    target_pass<T5C, H5C, W5C><<<NB_T5, 256, 0, stream>>>(
        cls5, reg5, tg5, mask5, ws + P_BBOX, ws + P_OBJP, ws + P_CLS, SLOT5);

    dense_pass<H4C, W4C><<<NB_D4, 256, 0, stream>>>(cls4, mask4, ws + P_NB4, ws + P_NC4);
    dense_pass<H5C, W5C><<<NB_D5, 256, 0, stream>>>(cls5, mask5, ws + P_NB5, ws + P_NC5);

    final_pass<<<1, 1, 0, stream>>>(ws, (float*)d_out);
}